// GlobalHistogramFixedSamplesAB_56727928045825
// MI455X (gfx1250) — compile-verified
//
#include <hip/hip_runtime.h>
#include <math.h>

// ---- problem constants (from reference) ----
#define BATCH 8
#define NVERT 8192
#define NSAMP 4096
#define TILES_PER_BATCH (NSAMP / 16)          // 256
#define WAVES_PER_BLOCK 8

static constexpr float K_BD = 1.0f / 63.0f;          // (1-0)/(64-1)
// C = 8192^(-1/6);  CF = (2pi)^-1 * C^-2 = 2^(13/3)/(2pi);  CN = -1/(2C^2) = -2^(13/3)/2
static constexpr float K_CF = 3.2083626f;
static constexpr float K_CN = -10.0793684f;
static constexpr float K_LOG2E = 1.44269504088896340736f;

typedef __attribute__((ext_vector_type(2))) float v2f;
typedef __attribute__((ext_vector_type(8))) float v8f;

// ---------------- kernel 1: per-batch sigma (unbiased std, mean over xy, clip) -------------
__global__ void sigma_kernel(const float* __restrict__ T, float* __restrict__ sigma_out) {
  __shared__ float sm[256];
  const int b = blockIdx.x;
  const int tid = threadIdx.x;
  const float2* Tb = (const float2*)T + (size_t)b * NVERT;
  float sx = 0.f, sy = 0.f, sxx = 0.f, syy = 0.f;
  for (int j = tid; j < NVERT; j += 256) {
    float2 t = Tb[j];
    sx += t.x; sy += t.y;
    sxx += t.x * t.x; syy += t.y * t.y;
  }
  float vals[4] = {sx, sy, sxx, syy};
  float red[4];
  for (int k = 0; k < 4; ++k) {
    sm[tid] = vals[k];
    __syncthreads();
    for (int off = 128; off > 0; off >>= 1) {
      if (tid < off) sm[tid] += sm[tid + off];
      __syncthreads();
    }
    red[k] = sm[0];
    __syncthreads();
  }
  if (tid == 0) {
    const float n = (float)NVERT;
    float vx = (red[2] - red[0] * red[0] / n) / (n - 1.0f);
    float vy = (red[3] - red[1] * red[1] / n) / (n - 1.0f);
    float sig = 0.5f * (sqrtf(fmaxf(vx, 0.f)) + sqrtf(fmaxf(vy, 0.f)));
    sigma_out[b] = fmaxf(sig, K_BD);
  }
}

// ---------------- kernel 2: WMMA KDE -> unnormalized K_H in d_out ---------------------------
// One wave computes one 16-row tile of K_H for one batch, sweeping all 8192 vertices
// in 16-column chunks. Augmented-K trick gives the 16x16 squared-distance tile from
// ONE v_wmma_f32_16x16x4_f32, and the per-batch exp2 scale (CN/sigma^2*log2e) is folded
// into the loop-invariant A operand, so the WMMA output feeds v_exp_f32 directly:
// loop body = build B + 1 wmma + 8 v_exp_f32 + 4 v_pk_add_f32 (trans-pipe bound).
__global__ void kde_kernel(const float* __restrict__ T, const float* __restrict__ S,
                           const float* __restrict__ sigma_arr, float* __restrict__ KH) {
  const int lane = threadIdx.x & 31;
  const int wave = threadIdx.x >> 5;
  const int tile = blockIdx.x * WAVES_PER_BLOCK + wave;   // 0 .. B*256-1
  const int b = tile / TILES_PER_BATCH;
  const int i0 = (tile % TILES_PER_BATCH) * 16;

  const float sigma = sigma_arr[b];
  const float inv_s2 = 1.0f / (sigma * sigma);
  const float k2 = K_CN * inv_s2 * K_LOG2E;      // exp2 exponent scale, folded into A

  const int m = lane & 15;

  // A operand (16x4 f32), pre-scaled by k2: lanes 0-15 hold K=0,1 ; lanes 16-31 hold K=2,3
  const float2 sv = ((const float2*)S)[(size_t)b * NSAMP + i0 + m];
  v2f a;
  if (lane < 16) { a.x = k2 * sv.x;                            a.y = k2 * sv.y; }  // K0,K1
  else           { a.x = k2 * (sv.x * sv.x + sv.y * sv.y);     a.y = k2;        }  // K2=k2|s|^2, K3=k2

  v8f acc = {0.f, 0.f, 0.f, 0.f, 0.f, 0.f, 0.f, 0.f};
  const float2* Tb = (const float2*)T + (size_t)b * NVERT;

#pragma unroll 2
  for (int j0 = 0; j0 < NVERT; j0 += 16) {
    const float2 tv = Tb[j0 + m];
    // B operand (4x16 f32): VGPR0 = rows K0 / K2 in lane halves, VGPR1 = rows K1 / K3
    v2f bb;
    if (lane < 16) { bb.x = -2.0f * tv.x; bb.y = -2.0f * tv.y; }              // K0,K1
    else           { bb.x = 1.0f;         bb.y = tv.x * tv.x + tv.y * tv.y; } // K2,K3

    v8f czero = {0.f, 0.f, 0.f, 0.f, 0.f, 0.f, 0.f, 0.f};
    // D[i][j] = k2 * ||S_i - T_j||^2  (16x16 tile, one instruction, exp2-ready)
    v8f d = __builtin_amdgcn_wmma_f32_16x16x4_f32(
        /*neg_a=*/false, a, /*neg_b=*/false, bb,
        /*c_mod=*/(short)0, czero, /*reuse_a=*/false, /*reuse_b=*/false);

#pragma unroll
    for (int v = 0; v < 8; ++v)
      acc[v] += __builtin_amdgcn_exp2f(d[v]);    // bare v_exp_f32, no extra muls
  }

  // Row sums: reduce each VGPR across its 16-lane half (masks 1,2,4,8 stay in-half).
#pragma unroll
  for (int v = 0; v < 8; ++v) {
    float s = acc[v];
    s += __shfl_xor(s, 1, 32);
    s += __shfl_xor(s, 2, 32);
    s += __shfl_xor(s, 4, 32);
    s += __shfl_xor(s, 8, 32);
    acc[v] = s;
  }

  const float outer = K_CF * inv_s2 * (1.0f / (float)NVERT);
  if (lane == 0 || lane == 16) {
    // lane 0: rows M=0..7 (VGPR v = row v); lane 16: rows M=8..15 (VGPR v = row v+8)
    const int base = b * NSAMP + i0 + (lane >> 4) * 8;
#pragma unroll
    for (int v = 0; v < 8; ++v)
      KH[base + v] = outer * acc[v];
  }
}

// ---------------- kernel 3: per-batch sum of K_H (deterministic tree) ----------------------
__global__ void sum_kernel(const float* __restrict__ KH, float* __restrict__ sums) {
  __shared__ float sm[256];
  const int b = blockIdx.x;
  const int tid = threadIdx.x;
  float s = 0.f;
  for (int i = tid; i < NSAMP; i += 256) s += KH[(size_t)b * NSAMP + i];
  sm[tid] = s;
  __syncthreads();
  for (int off = 128; off > 0; off >>= 1) {
    if (tid < off) sm[tid] += sm[tid + off];
    __syncthreads();
  }
  if (tid == 0) sums[b] = fmaxf(sm[0], 1e-5f);   // jnp.clip(sum, 1e-5) lower bound
}

// ---------------- kernel 4: in-place normalize ---------------------------------------------
__global__ void norm_kernel(float* __restrict__ out, const float* __restrict__ sums) {
  const int idx = blockIdx.x * blockDim.x + threadIdx.x;
  if (idx < BATCH * NSAMP)
    out[idx] = out[idx] / sums[idx / NSAMP];
}

extern "C" void kernel_launch(void* const* d_in, const int* in_sizes, int n_in,
                              void* d_out, int out_size, void* d_ws, size_t ws_size,
                              hipStream_t stream) {
  const float* T = (const float*)d_in[0];   // [B, NV, 2] f32
  const float* S = (const float*)d_in[1];   // [B, NS, 2] f32
  float* out = (float*)d_out;               // [B, NS] f32 (K_H scratch, then normalized in place)
  float* w = (float*)d_ws;
  float* sigma = w;        // 8 floats
  float* sums  = w + 8;    // 8 floats

  sigma_kernel<<<BATCH, 256, 0, stream>>>(T, sigma);

  const int tiles = BATCH * TILES_PER_BATCH;                 // 2048 waves
  kde_kernel<<<tiles / WAVES_PER_BLOCK, WAVES_PER_BLOCK * 32, 0, stream>>>(T, S, sigma, out);

  sum_kernel<<<BATCH, 256, 0, stream>>>(out, sums);

  norm_kernel<<<(BATCH * NSAMP + 255) / 256, 256, 0, stream>>>(out, sums);
}